// DeformationGrid_67791763800154
// MI455X (gfx1250) — compile-verified
//
#include <hip/hip_runtime.h>
#include <stdint.h>

// Trilinear interpolation of a (160,160,160,3) fp32 grid at N random points.
// Memory-bound gather kernel for MI455X (gfx1250):
//   - grid (~49 MB) stays L2-resident (192 MB L2): gathers use default RT hint,
//     8 corner gathers = global_load_b96 with immediate offsets off ONE base
//   - coord stream double-buffered into LDS: each lane async-loads its own
//     point's 12 B (+4 B slack) via global_load_async_to_lds_b128 into a
//     16 B-aligned private slot; consumed next iteration with an explicit
//     ds_load_b128 (DS pipe, bank-conflict-free). s_wait_asynccnt 0x1 keeps
//     the next chunk in flight while the current one is consumed.
//   - output is a write-once stream: nontemporal b96 stores keep L2 for grid
//   - wave32: 256-thread blocks = 8 waves; per-wave-private LDS slots -> no barriers

#define GNX 160
#define GNY 160
#define GNZ 160

#define LANE_SLOT   16                       // bytes per lane (float4)
#define WAVE_BYTES  (32 * LANE_SLOT)         // 512 B per wave chunk
#define BUF_BYTES   (8 * WAVE_BYTES)         // 4 KB per buffer (8 waves)

__global__ __launch_bounds__(256) void trilerp_async_kernel(
    const float* __restrict__ coords,  // [N,3]
    const float* __restrict__ grid,    // [160,160,160,3]
    float* __restrict__ out,           // [N,3]
    int n, int stride)
{
    // [buffer][wave][lane] float4 coord staging: 8 KB LDS, 16B slots.
    // Accessed exclusively through explicit LDS byte offsets (async write via
    // TDM-side engine, read via ds_load_b128 inline asm).
    __shared__ float4 sc[2][8][32];

    const int tid  = threadIdx.x;
    const int lane = tid & 31;
    const int wave = tid >> 5;

    // LDS byte offset of sc (low 32 bits of the generic address of an LDS
    // object are the LDS byte offset).
    const unsigned scb      = (unsigned)(uintptr_t)sc;
    const unsigned laneBase = scb + (unsigned)(wave * WAVE_BYTES + lane * LANE_SLOT);

    const float sx = (float)(GNX - 1);
    const float sy = (float)(GNY - 1);
    const float sz = (float)(GNZ - 1);

    const int i0 = blockIdx.x * 256 + tid;

    // Prologue: async-stage iteration 0's chunk. Each lane loads its own
    // point's coords (16 B from a 4-aligned address; DWORD-mode legal).
    // Guard +33 keeps lane 31's 4 B slack in-bounds; wave-uniform.
    {
        int iw = i0 - lane;                       // 32-aligned wave chunk base
        if (iw + 33 <= n) {
            const float* ga = coords + 3 * (size_t)i0;
            asm volatile("global_load_async_to_lds_b128 %0, %1, off"
                         :: "v"(laneBase), "v"(ga) : "memory");
        }
    }

    int buf = 0;
    for (int i = i0; i < n; i += stride) {
        const int  iw    = i - lane;              // wave-uniform chunk base
        const bool fullC = (iw + 33) <= n;        // wave-uniform: chunk staged?
        const int  iwn   = iw + stride;
        const bool fullN = (iwn + 33) <= n;       // wave-uniform: stage next?

        // Issue next iteration's chunk into the other buffer (1 async op in
        // flight while we consume the current chunk).
        if (fullN) {
            const float* ga = coords + 3 * (size_t)(i + stride);
            unsigned ldsa   = laneBase + (unsigned)((buf ^ 1) * BUF_BYTES);
            asm volatile("global_load_async_to_lds_b128 %0, %1, off"
                         :: "v"(ldsa), "v"(ga) : "memory");
        }

        // Get this point's coords.
        float cx, cy, cz;
        if (fullC) {
            // Drain the CURRENT chunk's async op; the NEXT chunk's op stays
            // outstanding (async loads complete in order).
            if (fullN) asm volatile("s_wait_asynccnt 0x1" ::: "memory");
            else       asm volatile("s_wait_asynccnt 0x0" ::: "memory");
            // Explicit DS-pipe read of this lane's 16 B slot (align 16,
            // bank-conflict-free). s_wait_dscnt inside the asm because the
            // compiler cannot track DScnt for instructions it did not emit.
            float4 c4;
            unsigned raddr = laneBase + (unsigned)(buf * BUF_BYTES);
            asm volatile("ds_load_b128 %0, %1\n\t"
                         "s_wait_dscnt 0x0"
                         : "=v"(c4) : "v"(raddr) : "memory");
            cx = c4.x; cy = c4.y; cz = c4.z;      // .w = slack, ignored
        } else {
            // Edge (last chunk) fallback: direct nontemporal loads.
            const float* cp = coords + 3 * (size_t)i;
            cx = __builtin_nontemporal_load(cp + 0);
            cy = __builtin_nontemporal_load(cp + 1);
            cz = __builtin_nontemporal_load(cp + 2);
        }

        // Voxel-space position, faithful to map_coordinates(order=1, cval=0).
        float px = cx * sx, py = cy * sy, pz = cz * sz;
        float fx = floorf(px), fy = floorf(py), fz = floorf(pz);
        float tx = px - fx,    ty = py - fy,    tz = pz - fz;
        int ix = (int)fx, iy = (int)fy, iz = (int)fz;

        // One base address for the (0,0,0) corner; clamp keeps every immediate
        // corner offset inside the grid. For coords in [0,1) (the reference's
        // input domain), floor() is already in [0,158], so this is exact and
        // all 8 corners are in-range (the reference's cval=0 masks are all
        // true); the clamp just guarantees memory safety for any input.
        int bx = min(max(ix, 0), GNX - 2);
        int by = min(max(iy, 0), GNY - 2);
        int bz = min(max(iz, 0), GNZ - 2);
        const float* base = grid + 3 * (size_t)((bx * GNY + by) * GNZ + bz);

        // Phase 1: issue all 8 corner gathers (b96, compile-time immediate
        // offsets off one VGPR address pair) before any consumption.
        float g[8][3];
#pragma unroll
        for (int c = 0; c < 8; ++c) {
            const int dx = (c >> 2) & 1, dy = (c >> 1) & 1, dz = c & 1;
            const float* p = base + dx * (3 * GNY * GNZ) + dy * (3 * GNZ) + dz * 3;
            g[c][0] = p[0];
            g[c][1] = p[1];
            g[c][2] = p[2];
        }

        // Phase 2: corner weights (independent ALU overlaps the gathers).
        float wxa[2] = {1.0f - tx, tx};
        float wya[2] = {1.0f - ty, ty};
        float wza[2] = {1.0f - tz, tz};
        float w[8];
#pragma unroll
        for (int c = 0; c < 8; ++c) {
            const int dx = (c >> 2) & 1, dy = (c >> 1) & 1, dz = c & 1;
            w[c] = (wxa[dx] * wya[dy]) * wza[dz];
        }

        // Phase 3: accumulate in reference corner order (dz fastest).
        float a0 = 0.0f, a1 = 0.0f, a2 = 0.0f;
#pragma unroll
        for (int c = 0; c < 8; ++c) {
            a0 += w[c] * g[c][0];
            a1 += w[c] * g[c][1];
            a2 += w[c] * g[c][2];
        }

        // Write-once stream: nontemporal b96 store.
        float* op = out + 3 * (size_t)i;
        __builtin_nontemporal_store(a0, op + 0);
        __builtin_nontemporal_store(a1, op + 1);
        __builtin_nontemporal_store(a2, op + 2);

        buf ^= 1;
    }
}

extern "C" void kernel_launch(void* const* d_in, const int* in_sizes, int n_in,
                              void* d_out, int out_size, void* d_ws, size_t ws_size,
                              hipStream_t stream) {
    const float* coords = (const float*)d_in[0];   // [N,3] fp32
    const float* grid   = (const float*)d_in[1];   // [160,160,160,3] fp32
    float* out = (float*)d_out;                    // [N,3] fp32

    int n = in_sizes[0] / 3;                       // N_POINTS

    const int block = 256;                         // 8 wave32 waves
    int blocks = (n + block - 1) / block;
    const int maxBlocks = 8192;                    // grid-stride: 2 pts/thread at N=4.19M
    if (blocks > maxBlocks) blocks = maxBlocks;
    int stride = blocks * block;

    trilerp_async_kernel<<<blocks, block, 0, stream>>>(coords, grid, out, n, stride);
}